// RoIHeads_9835475108018
// MI455X (gfx1250) — compile-verified
//
#include <hip/hip_runtime.h>
#include <stdint.h>

// ---------------- problem constants ----------------
#define BATCH 4
#define NPROP 8000
#define NCLS  91
#define NFG   (NCLS - 1)          // foreground classes
#define CAP   32768               // per-image candidate capacity (256KB LDS sort)
#define TOPK  1000                // PRE_NMS_TOPK
#define KP    1008                // TOPK padded to 63 * 16 for WMMA tiles
#define TILES 63
#define DETS  100

static constexpr float SCORE_TH  = 0.05f;
static constexpr float NMS_TH    = 0.5f;
static constexpr float DW_CLIP_F = 4.135166556742356f;   // log(1000/16)

// ---------------- workspace layout (bytes) ----------------
static constexpr size_t OFF_CNT    = 0;                                   // BATCH ints
static constexpr size_t OFF_KEYS   = 256;                                 // BATCH*CAP u64
static constexpr size_t KEYS_BYTES = (size_t)BATCH * CAP * 8;             // 1,048,576
static constexpr size_t OFF_OBOX   = OFF_KEYS + KEYS_BYTES;               // offset boxes  BATCH*KP*4 f32
static constexpr size_t OBOX_BYTES = (size_t)BATCH * KP * 4 * 4;          // 64,512
static constexpr size_t OFF_BOX    = OFF_OBOX + OBOX_BYTES;               // clean boxes   BATCH*KP*4 f32
static constexpr size_t OFF_SC     = OFF_BOX + OBOX_BYTES;                // BATCH*TOPK f32
static constexpr size_t SC_BYTES   = (size_t)BATCH * TOPK * 4;
static constexpr size_t OFF_LB     = OFF_SC + SC_BYTES;                   // BATCH*TOPK i32
static constexpr size_t OFF_IOU    = ((OFF_LB + SC_BYTES + 255) / 256) * 256; // BATCH*KP*KP f32

typedef __attribute__((ext_vector_type(2))) float v2f;
typedef __attribute__((ext_vector_type(8))) float v8f;

// ---------------- shared decode helper (BoxCoder.decode + clip) ----------------
__device__ __forceinline__ void decode_clip(const float* __restrict__ r4,
                                            float px1, float py1, float px2, float py2,
                                            float Wf, float Hf,
                                            float& x1, float& y1, float& x2, float& y2) {
    float w  = px2 - px1, h = py2 - py1;
    float cx = px1 + 0.5f * w, cy = py1 + 0.5f * h;
    float dx = r4[0] * 0.1f;                       // / WX
    float dy = r4[1] * 0.1f;                       // / WY
    float dw = fminf(r4[2] * 0.2f, DW_CLIP_F);     // / WW, clipped
    float dh = fminf(r4[3] * 0.2f, DW_CLIP_F);     // / WH, clipped
    float pcx = dx * w + cx, pcy = dy * h + cy;
    float pw  = expf(dw) * w, ph = expf(dh) * h;
    x1 = fminf(fmaxf(pcx - 0.5f * pw, 0.f), Wf);
    y1 = fminf(fmaxf(pcy - 0.5f * ph, 0.f), Hf);
    x2 = fminf(fmaxf(pcx + 0.5f * pw, 0.f), Wf);
    y2 = fminf(fmaxf(pcy + 0.5f * ph, 0.f), Hf);
}

// ---------------- kernel 0: zero counters + output ----------------
__global__ void init_kernel(int* __restrict__ counters, float* __restrict__ out, int outN) {
    int t = blockIdx.x * blockDim.x + threadIdx.x;
    if (t < BATCH) counters[t] = 0;
    if (t < outN)  out[t] = 0.f;
}

// ---------------- kernel 1: softmax + filter + key append ----------------
// One wave32 per (b, n) row. Only survivors (~2%) touch box_regression.
__global__ void score_filter_kernel(const float* __restrict__ logits,
                                    const float* __restrict__ reg,
                                    const float* __restrict__ props,
                                    const int*   __restrict__ imghw,
                                    int* __restrict__ counters,
                                    unsigned long long* __restrict__ keys) {
    int gwave = (blockIdx.x * blockDim.x + threadIdx.x) >> 5;
    int lane  = threadIdx.x & 31;
    if (gwave >= BATCH * NPROP) return;
    int b = gwave / NPROP, n = gwave % NPROP;

    const float* lrow = logits + (size_t)gwave * NCLS;
    __builtin_prefetch(lrow, 0, 0);                     // global_prefetch_b8

    float v0 = lrow[lane];
    float v1 = lrow[lane + 32];
    float v2 = (lane + 64 < NCLS) ? lrow[lane + 64] : -3.0e38f;
    float m = fmaxf(v0, fmaxf(v1, v2));
    #pragma unroll
    for (int s = 16; s > 0; s >>= 1) m = fmaxf(m, __shfl_xor(m, s, 32));
    float e = expf(v0 - m) + expf(v1 - m) + ((lane + 64 < NCLS) ? expf(v2 - m) : 0.f);
    #pragma unroll
    for (int s = 16; s > 0; s >>= 1) e += __shfl_xor(e, s, 32);
    float inv = 1.f / e;

    const float* p4 = props + ((size_t)b * NPROP + n) * 4;
    float px1 = p4[0], py1 = p4[1], px2 = p4[2], py2 = p4[3];
    float Hf = (float)imghw[b * 2], Wf = (float)imghw[b * 2 + 1];

    for (int c = 1 + lane; c < NCLS; c += 32) {          // foreground classes only
        float sc = expf(lrow[c] - m) * inv;
        if (sc > SCORE_TH) {
            const float* r4 = reg + ((size_t)gwave * NCLS + c) * 4;
            float x1, y1, x2, y2;
            decode_clip(r4, px1, py1, px2, py2, Wf, Hf, x1, y1, x2, y2);
            if ((x2 - x1) >= 1.f && (y2 - y1) >= 1.f) {  // remove_small_boxes
                unsigned id = (unsigned)(n * NFG + (c - 1));      // flat cand index < 2^20
                unsigned long long key =
                    ((unsigned long long)__float_as_uint(sc) << 20)
                    | (unsigned long long)(0xFFFFFu - id);        // tie-break: lower id wins
                int slot = atomicAdd(&counters[b], 1);
                if (slot < CAP) keys[(size_t)b * CAP + slot] = key;
            }
        }
    }
}

// ---------------- kernel 2: 256KB-LDS bitonic sort (descending) + finalist decode ----------------
__global__ void sort_select_kernel(const float* __restrict__ reg,
                                   const float* __restrict__ props,
                                   const int*   __restrict__ imghw,
                                   const int*   __restrict__ counters,
                                   const unsigned long long* __restrict__ keysIn,
                                   float* __restrict__ boxes,     // clean, [B][KP][4]
                                   float* __restrict__ oboxes,    // +class offset, [B][KP][4]
                                   float* __restrict__ scores,    // [B][TOPK]
                                   int*   __restrict__ labels) {  // [B][TOPK]
    extern __shared__ unsigned long long sk[];   // CAP u64 = 256 KB (CDNA5: 320KB/WGP)
    int b = blockIdx.x, tid = threadIdx.x;
    int cnt = counters[b]; if (cnt > CAP) cnt = CAP;
    for (int i = tid; i < CAP; i += blockDim.x)
        sk[i] = (i < cnt) ? keysIn[(size_t)b * CAP + i] : 0ull;
    __syncthreads();

    for (int k = 2; k <= CAP; k <<= 1) {
        for (int j = k >> 1; j > 0; j >>= 1) {
            for (int i = tid; i < CAP; i += blockDim.x) {
                int ixj = i ^ j;
                if (ixj > i) {
                    bool up = (i & k) == 0;                 // descending blocks
                    unsigned long long a = sk[i], c = sk[ixj];
                    if (up ? (a < c) : (a > c)) { sk[i] = c; sk[ixj] = a; }
                }
            }
            __syncthreads();
        }
    }

    float Hf = (float)imghw[b * 2], Wf = (float)imghw[b * 2 + 1];
    float off_scale = fmaxf(Hf, Wf) + 2.f;                  // batched_nms offset trick
    for (int t = tid; t < KP; t += blockDim.x) {
        float x1 = 0.f, y1 = 0.f, x2 = 0.f, y2 = 0.f, sc = 0.f, of = 0.f;
        int lb = 0;
        if (t < TOPK) {
            unsigned long long key = sk[t];
            if (key != 0ull) {
                sc = __uint_as_float((unsigned)(key >> 20));
                unsigned id = 0xFFFFFu - (unsigned)(key & 0xFFFFFull);
                int n = (int)(id / NFG);
                int c = (int)(id % NFG) + 1;
                const float* p4 = props + ((size_t)b * NPROP + n) * 4;
                const float* r4 = reg + (((size_t)b * NPROP + n) * NCLS + c) * 4;
                decode_clip(r4, p4[0], p4[1], p4[2], p4[3], Wf, Hf, x1, y1, x2, y2);
                lb = c;
                of = (float)c * off_scale;
            }
        }
        size_t base = ((size_t)b * KP + t) * 4;
        boxes[base + 0] = x1;      boxes[base + 1] = y1;
        boxes[base + 2] = x2;      boxes[base + 3] = y2;
        oboxes[base + 0] = x1 + of; oboxes[base + 1] = y1 + of;
        oboxes[base + 2] = x2 + of; oboxes[base + 3] = y2 + of;
        if (t < TOPK) { scores[b * TOPK + t] = sc; labels[b * TOPK + t] = lb; }
    }
}

// ---------------- kernel 3: IoU 16x16 tiles; union denominator via WMMA ----------------
// D[M][N] = area_i[M]*1 + 1*area_j[N]  computed by V_WMMA_F32_16X16X4_F32
// A (16x4): row M = [area_i, 1, 0, 0];  B (4x16): col N = [1, area_j, 0, 0]^T
__global__ void iou_tile_kernel(const float* __restrict__ oboxes,
                                float* __restrict__ iou) {
    int gw   = (blockIdx.x * blockDim.x + threadIdx.x) >> 5;   // uniform per wave
    int lane = threadIdx.x & 31;
    if (gw >= BATCH * TILES * TILES) return;
    int b   = gw / (TILES * TILES);
    int rem = gw % (TILES * TILES);
    int tm  = rem / TILES, tn = rem % TILES;
    const float* ob = oboxes + (size_t)b * KP * 4;
    int l15 = lane & 15;

    v2f A  = {0.f, 0.f};     // lanes 0-15: (K=0,K=1); lanes 16-31: (K=2,K=3) = 0
    v2f Bm = {0.f, 0.f};
    if (lane < 16) {
        const float* bi = ob + (size_t)(tm * 16 + lane) * 4;
        A.x = (bi[2] - bi[0]) * (bi[3] - bi[1]);   // area_i[M]
        A.y = 1.0f;
        const float* bj = ob + (size_t)(tn * 16 + lane) * 4;
        Bm.x = 1.0f;
        Bm.y = (bj[2] - bj[0]) * (bj[3] - bj[1]);  // area_j[N]
    }
    v8f Cz = {};
    v8f D = __builtin_amdgcn_wmma_f32_16x16x4_f32(false, A, false, Bm,
                                                  (short)0, Cz, false, false);

    const float* bj = ob + (size_t)(tn * 16 + l15) * 4;        // column box, all lanes
    float jx1 = bj[0], jy1 = bj[1], jx2 = bj[2], jy2 = bj[3];
    int mbase = tm * 16 + ((lane < 16) ? 0 : 8);               // C/D layout: M = r or r+8
    float* orow = iou + (size_t)b * KP * KP;
    #pragma unroll
    for (int r = 0; r < 8; ++r) {
        const float* bi = ob + (size_t)(mbase + r) * 4;
        float iw = fmaxf(fminf(bi[2], jx2) - fmaxf(bi[0], jx1), 0.f);
        float ih = fmaxf(fminf(bi[3], jy2) - fmaxf(bi[1], jy1), 0.f);
        float inter = iw * ih;
        orow[(size_t)(mbase + r) * KP + (tn * 16 + l15)] =
            inter / (D[r] - inter + 1e-9f);                    // union = D - inter
    }
}

// ---------------- kernel 4: greedy NMS scan + first-100-kept emit ----------------
__global__ void nms_out_kernel(const float* __restrict__ boxes,
                               const float* __restrict__ scores,
                               const int*   __restrict__ labels,
                               const float* __restrict__ iou,
                               float* __restrict__ out) {
    __shared__ float ssc[TOPK];
    __shared__ int   supp[TOPK];
    __shared__ int   pfx[1024];
    int b = blockIdx.x, tid = threadIdx.x;
    if (tid < TOPK) { ssc[tid] = scores[b * TOPK + tid]; supp[tid] = 0; }
    __syncthreads();

    const float* imat = iou + (size_t)b * KP * KP;
    for (int i = 0; i < TOPK; ++i) {
        bool alive = (ssc[i] > 0.f) && (supp[i] == 0);         // uniform across block
        if (tid < TOPK && i + 1 < TOPK)
            __builtin_prefetch(&imat[(size_t)(i + 1) * KP + tid], 0, 0);
        if (alive && tid > i && tid < TOPK) {
            if (imat[(size_t)i * KP + tid] > NMS_TH) supp[tid] = 1;
        }
        __syncthreads();
    }

    int keep = (tid < TOPK && ssc[tid] > 0.f && supp[tid] == 0) ? 1 : 0;
    pfx[tid] = keep;
    __syncthreads();
    for (int d = 1; d < 1024; d <<= 1) {                       // Hillis-Steele scan
        int v = (tid >= d) ? pfx[tid - d] : 0;
        __syncthreads();
        pfx[tid] += v;
        __syncthreads();
    }
    int rank = pfx[tid] - keep;                                // exclusive rank among kept
    if (keep && rank < DETS) {                                 // already score-sorted
        const float* bx = boxes + ((size_t)b * KP + tid) * 4;
        float* ob = out + ((size_t)b * DETS + rank) * 4;
        ob[0] = bx[0]; ob[1] = bx[1]; ob[2] = bx[2]; ob[3] = bx[3];
        out[BATCH * DETS * 4 + b * DETS + rank] = ssc[tid];
        out[BATCH * DETS * 4 + BATCH * DETS + b * DETS + rank] =
            (float)labels[b * TOPK + tid];
    }
}

// ---------------- launch ----------------
extern "C" void kernel_launch(void* const* d_in, const int* in_sizes, int n_in,
                              void* d_out, int out_size, void* d_ws, size_t ws_size,
                              hipStream_t stream) {
    const float* logits = (const float*)d_in[0];
    const float* reg    = (const float*)d_in[1];
    const float* props  = (const float*)d_in[2];
    const int*   imghw  = (const int*)d_in[3];
    float* out = (float*)d_out;
    char*  ws  = (char*)d_ws;

    int* counters               = (int*)(ws + OFF_CNT);
    unsigned long long* keys    = (unsigned long long*)(ws + OFF_KEYS);
    float* oboxes               = (float*)(ws + OFF_OBOX);
    float* boxes                = (float*)(ws + OFF_BOX);
    float* scores               = (float*)(ws + OFF_SC);
    int*   labels               = (int*)(ws + OFF_LB);
    float* iou                  = (float*)(ws + OFF_IOU);

    // 256 KB dynamic LDS for the sort (legal on CDNA5's 320KB/WGP; opt-in attr)
    (void)hipFuncSetAttribute((const void*)sort_select_kernel,
                              hipFuncAttributeMaxDynamicSharedMemorySize,
                              CAP * (int)sizeof(unsigned long long));

    init_kernel<<<(out_size + 255) / 256 + 1, 256, 0, stream>>>(counters, out, out_size);

    int rows = BATCH * NPROP;                              // one wave32 per row
    score_filter_kernel<<<(rows * 32 + 255) / 256, 256, 0, stream>>>(
        logits, reg, props, imghw, counters, keys);

    sort_select_kernel<<<BATCH, 1024, CAP * sizeof(unsigned long long), stream>>>(
        reg, props, imghw, counters, keys, boxes, oboxes, scores, labels);

    int tw = BATCH * TILES * TILES;                        // one wave32 per 16x16 tile
    iou_tile_kernel<<<(tw * 32 + 255) / 256, 256, 0, stream>>>(oboxes, iou);

    nms_out_kernel<<<BATCH, 1024, 0, stream>>>(boxes, scores, labels, iou, out);

    (void)in_sizes; (void)n_in; (void)ws_size;
}